// DConv_78374563218112
// MI455X (gfx1250) — compile-verified
//
#include <hip/hip_runtime.h>
#include <hip/hip_bf16.h>
#include <stddef.h>
#include <stdint.h>

// Deformable conv v2 (DCNv2) for MI455X / gfx1250.
// Fused implicit GEMM: never materialize the 151MB column tensor v.
// - per-tap 128x16 v-tiles built in LDS from bilinear gathers (NHWC x, L2-resident)
// - per-tap 128x128 weight tiles DMA'd into LDS with global_load_async_to_lds_b128
//   (ASYNCcnt path), overlapping the gather VALU work
// - contraction on V_WMMA_F32_16X16X4_F32: f32 tensor path == f32 reference numerics

#define BB 8
#define CC 128
#define HH 64
#define WW 64
#define OO 128
#define KK 9
#define HWSZ 4096                  // H*W
#define KTOT 1152                  // C*KK
#define P_TOT (BB * KK * HWSZ)     // 294912 param entries

// workspace layout (in floats)
#define OFF_XT    0                       // x transposed to NHWC: 8*64*64*128
#define OFF_WORD  4194304                 // weight reordered [O][k*128+c]: 128*1152
#define OFF_PC    (OFF_WORD + 147456)     // packed corner coords (int): P_TOT
#define OFF_PW    (OFF_PC + P_TOT)        // 4 planes of bilinear weights: 4*P_TOT
// total ~ 5.82M floats ~ 23.3 MB of workspace

#define WT_STRIDE 132                     // wtile row stride (dwords); 132%64==4 ->
                                          // b64 A-frag reads cover all 64 banks once

typedef float v2f __attribute__((ext_vector_type(2)));
typedef float v8f __attribute__((ext_vector_type(8)));

// ------------------------------------------- prep: NCHW -> NHWC (LDS-tiled, coalesced)
// block: 256 threads, tile = 64 spatial positions x 128 channels
__global__ __launch_bounds__(256) void transpose_x_kernel(
    const float* __restrict__ x, float* __restrict__ xT)
{
    __shared__ float tile[CC * 65];               // [c][hw], pad 65 vs banks
    const int tid = threadIdx.x;
    const int b   = blockIdx.x >> 6;              // 64 tiles per image
    const int hw0 = (blockIdx.x & 63) << 6;       // 64 consecutive hw positions

    const float* xb = x + ((size_t)b << 19);      // b*C*H*W
#pragma unroll
    for (int j = 0; j < 32; ++j) {                // read: coalesced over hw
        const int linear = (j << 8) + tid;        // 0..8191
        const int c  = linear >> 6;
        const int hw = linear & 63;
        tile[c * 65 + hw] = xb[((size_t)c << 12) + hw0 + hw];
    }
    __syncthreads();
    float* xTb = xT + ((size_t)b << 19) + ((size_t)hw0 << 7);
#pragma unroll
    for (int j = 0; j < 32; ++j) {                // write: coalesced over c
        const int linear = (j << 8) + tid;
        const int hw = linear >> 7;
        const int c  = linear & 127;
        xTb[((size_t)hw << 7) + c] = tile[c * 65 + hw];
    }
}

// ------------------------------------------- prep: weight [O][c*9+k] -> [O][k*128+c]
__global__ __launch_bounds__(256) void reorder_w_kernel(
    const float* __restrict__ w, float* __restrict__ wOrd)
{
    int idx = blockIdx.x * 256 + threadIdx.x;     // 128*1152 = 147456
    if (idx >= OO * KTOT) return;
    int o = idx / KTOT;
    int r = idx - o * KTOT;
    int k = r >> 7;
    int c = r & 127;
    wOrd[idx] = w[o * KTOT + c * KK + k];
}

// --------------------- offset/mask conv + bilinear parameter precomputation
// one thread per (pixel, tap): computes the 3 offset-conv channels {k, 9+k, 18+k}
// for its tap, then emits packed clamped corner coords + 4 premultiplied weights.
// k is uniform per block (idx = k*32768 + p) so w_off reads stay scalar.
__global__ __launch_bounds__(256) void dconv_offsets_kernel(
    const float* __restrict__ x,       // NCHW
    const float* __restrict__ w_off,   // [27][128][3][3]
    const float* __restrict__ b_off,   // [27]
    int*         __restrict__ pcoord,  // [B][KK][H][W] packed coords
    float*       __restrict__ pw)      // 4 planes [B][KK][H][W]
{
    const int idx = blockIdx.x * 256 + threadIdx.x;   // 0..294911
    const int k   = idx >> 15;                        // tap, uniform per block
    const int p   = idx & 32767;
    const int b   = p >> 12;
    const int rem = p & 4095;
    const int h   = rem >> 6;
    const int w   = rem & 63;

    float a0 = b_off[k];           // o1[k]  (y offset)
    float a1 = b_off[9 + k];       // o2[k]  (x offset)
    float a2 = b_off[18 + k];      // m[k]   (mask logit)

    const float* xb = x + ((size_t)b << 19);
    for (int kh = 0; kh < 3; ++kh) {
        const int y = h - 1 + kh;
        if ((unsigned)y >= (unsigned)HH) continue;
        for (int kw = 0; kw < 3; ++kw) {
            const int xx = w - 1 + kw;
            if ((unsigned)xx >= (unsigned)WW) continue;
            const float* xp = xb + (y << 6) + xx;              // + c*4096
            const float* w0 = w_off + (size_t)k * KTOT        + kh * 3 + kw;
            const float* w1 = w_off + (size_t)(9 + k) * KTOT  + kh * 3 + kw;
            const float* w2 = w_off + (size_t)(18 + k) * KTOT + kh * 3 + kw;
            for (int c = 0; c < CC; ++c) {
                const float xv = xp[(size_t)c << 12];
                a0 = fmaf(xv, w0[c * KK], a0);
                a1 = fmaf(xv, w1[c * KK], a1);
                a2 = fmaf(xv, w2[c * KK], a2);
            }
        }
    }

    const int ky = k / 3, kx = k % 3;
    const float py = (float)(h - 1 + ky) + a0;
    const float px = (float)(w - 1 + kx) + a1;
    const float msk = 1.0f / (1.0f + __expf(-a2));
    const float y0f = floorf(py), x0f = floorf(px);
    const float fy = py - y0f,  fx = px - x0f;
    const int y0 = (int)y0f, x0 = (int)x0f;
    const int y1 = y0 + 1,   x1 = x0 + 1;
    const float vy0 = (y0 >= 0 && y0 < HH) ? 1.f : 0.f;
    const float vy1 = (y1 >= 0 && y1 < HH) ? 1.f : 0.f;
    const float vx0 = (x0 >= 0 && x0 < WW) ? 1.f : 0.f;
    const float vx1 = (x1 >= 0 && x1 < WW) ? 1.f : 0.f;
    const int yc0 = min(max(y0, 0), HH - 1);
    const int yc1 = min(max(y1, 0), HH - 1);
    const int xc0 = min(max(x0, 0), WW - 1);
    const int xc1 = min(max(x1, 0), WW - 1);

    const int pidx = ((b * KK + k) << 12) + rem;
    pcoord[pidx] = yc0 | (xc0 << 8) | (yc1 << 16) | (xc1 << 24);
    pw[pidx]             = (1.f - fy) * (1.f - fx) * msk * vy0 * vx0;
    pw[P_TOT + pidx]     = (1.f - fy) * fx         * msk * vy0 * vx1;
    pw[2 * P_TOT + pidx] = fy         * (1.f - fx) * msk * vy1 * vx0;
    pw[3 * P_TOT + pidx] = fy         * fx         * msk * vy1 * vx1;
}

// ------------------------------------------------ fused implicit GEMM with WMMA
// block = 256 threads = 8 waves; tile = M(all 128 outputs) x N(16 spatial pos).
// per tap k:
//   1) issue async DMA of the 128x128 weight tile into LDS (ASYNCcnt)
//   2) build the 128x16 v-tile in LDS (bilinear gather, overlaps the DMA)
//   3) s_wait_asynccnt 0 + barrier, then 32 x V_WMMA_F32_16X16X4_F32, all-LDS fed
__global__ __launch_bounds__(256) void dconv_gemm_wmma_kernel(
    const float* __restrict__ xT,      // NHWC
    const float* __restrict__ wOrd,    // [O][k*128+c]
    const int*   __restrict__ pcoord,
    const float* __restrict__ pw,
    const float* __restrict__ bias,
    float*       __restrict__ out)     // NCHW [B][O][H][W]
{
    __shared__ float wtile[OO * WT_STRIDE];    // 67584 B, padded for conflict-free b64
    __shared__ float vtile[CC * 17];           // 8704 B, padded stride 17
    __shared__ int   sC[KK * 16];
    __shared__ float sW0[KK * 16], sW1[KK * 16], sW2[KK * 16], sW3[KK * 16];

    const int tid  = threadIdx.x;
    const int wave = tid >> 5;
    const int lane = tid & 31;

    const int n0  = blockIdx.x << 4;           // 16 consecutive pixels, same row
    const int b   = n0 >> 12;
    const int rem = n0 & 4095;
    const int h   = rem >> 6;
    const int w0c = rem & 63;

    if (tid < KK * 16) {
        const int pidx = ((b * KK + (tid >> 4)) << 12) + (h << 6) + w0c + (tid & 15);
        sC[tid]  = pcoord[pidx];
        sW0[tid] = pw[pidx];
        sW1[tid] = pw[P_TOT + pidx];
        sW2[tid] = pw[2 * P_TOT + pidx];
        sW3[tid] = pw[3 * P_TOT + pidx];
    }
    __syncthreads();

    v8f acc = {0.f, 0.f, 0.f, 0.f, 0.f, 0.f, 0.f, 0.f};

    const int mrow = (wave << 4) + (lane & 15);   // output-channel row of A frag
    const int ksel = (lane >> 4) << 1;            // lanes 0-15 -> K{0,1}; 16-31 -> K{2,3}
    const int ncol = lane & 15;

    const float* xTb = xT + ((size_t)b << 19);    // b*H*W*C
    const int sn = tid & 15;                      // staging: position within tile
    const int cb = (tid >> 4) << 3;               // staging: 8 channels per thread

    for (int k = 0; k < KK; ++k) {
        // ---- 1) async DMA: weight tile for tap k -> LDS (16B per lane per op)
        {
            const float* wk = wOrd + (k << 7);
#pragma unroll
            for (int i = 0; i < 16; ++i) {
                const int linear = (i << 8) + tid;           // 0..4095 float4 chunks
                const int o = linear >> 5;                   // 32 chunks per row
                const int c = (linear & 31) << 2;
                const uint64_t gaddr =
                    (uint64_t)(uintptr_t)(wk + (size_t)o * KTOT + c);
                const uint32_t laddr =
                    (uint32_t)(uintptr_t)(&wtile[o * WT_STRIDE + c]);
                asm volatile("global_load_async_to_lds_b128 %0, %1, off"
                             :: "v"(laddr), "v"(gaddr) : "memory");
            }
        }

        // ---- 2) build v tile for tap k (bilinear gather from NHWC x)
        {
            const int pi = (k << 4) + sn;
            const int pc = sC[pi];
            const int y0 =  pc        & 255;
            const int x0 = (pc >> 8)  & 255;
            const int y1 = (pc >> 16) & 255;
            const int x1 = (pc >> 24) & 255;
            const float w00 = sW0[pi], w01 = sW1[pi], w10 = sW2[pi], w11 = sW3[pi];
            const float4* p00 = (const float4*)(xTb + (((y0 << 6) + x0) << 7) + cb);
            const float4* p01 = (const float4*)(xTb + (((y0 << 6) + x1) << 7) + cb);
            const float4* p10 = (const float4*)(xTb + (((y1 << 6) + x0) << 7) + cb);
            const float4* p11 = (const float4*)(xTb + (((y1 << 6) + x1) << 7) + cb);
#pragma unroll
            for (int half = 0; half < 2; ++half) {
                const float4 a0 = p00[half], a1 = p01[half];
                const float4 a2 = p10[half], a3 = p11[half];
                const int c = cb + (half << 2);
                vtile[(c + 0) * 17 + sn] = w00 * a0.x + w01 * a1.x + w10 * a2.x + w11 * a3.x;
                vtile[(c + 1) * 17 + sn] = w00 * a0.y + w01 * a1.y + w10 * a2.y + w11 * a3.y;
                vtile[(c + 2) * 17 + sn] = w00 * a0.z + w01 * a1.z + w10 * a2.z + w11 * a3.z;
                vtile[(c + 3) * 17 + sn] = w00 * a0.w + w01 * a1.w + w10 * a2.w + w11 * a3.w;
            }
        }

        // ---- 3) all async weight DMA visible + all v-tile writes visible
        asm volatile("s_wait_asynccnt 0x0" ::: "memory");
        __syncthreads();

        // ---- 32 WMMA K-steps over the 128 channels of this tap, LDS-fed
        const float* wrow = wtile + mrow * WT_STRIDE + ksel;
#pragma unroll 4
        for (int c0 = 0; c0 < CC; c0 += 4) {
            v2f afrag = *(const v2f*)(wrow + c0);             // ds_load_b64
            v2f bfrag;
            bfrag.x = vtile[(c0 + ksel)     * 17 + ncol];     // ds_load_2addr
            bfrag.y = vtile[(c0 + ksel + 1) * 17 + ncol];
            acc = __builtin_amdgcn_wmma_f32_16x16x4_f32(
                false, afrag, false, bfrag, (short)0, acc, false, false);
        }
        __syncthreads();
    }

    // ---- epilogue: D layout = VGPR r: lanes0-15 M=r, lanes16-31 M=r+8
    const int obase = (wave << 4) + ((lane >> 4) << 3);
    float* outp = out + (((size_t)b * OO) << 12) + (h << 6) + w0c + ncol;
#pragma unroll
    for (int r = 0; r < 8; ++r) {
        const int o = obase + r;
        outp[(size_t)o << 12] = acc[r] + bias[o];
    }
}

extern "C" void kernel_launch(void* const* d_in, const int* in_sizes, int n_in,
                              void* d_out, int out_size, void* d_ws, size_t ws_size,
                              hipStream_t stream) {
    const float* x      = (const float*)d_in[0];
    const float* weight = (const float*)d_in[1];
    const float* bias   = (const float*)d_in[2];
    const float* w_off  = (const float*)d_in[3];
    const float* b_off  = (const float*)d_in[4];
    float* out = (float*)d_out;

    float* ws     = (float*)d_ws;        // needs ~24 MB of workspace
    float* xT     = ws + OFF_XT;
    float* wOrd   = ws + OFF_WORD;
    int*   pcoord = (int*)(ws + OFF_PC);
    float* pw     = ws + OFF_PW;

    transpose_x_kernel   <<<512,  256, 0, stream>>>(x, xT);
    reorder_w_kernel     <<<576,  256, 0, stream>>>(weight, wOrd);
    dconv_offsets_kernel <<<1152, 256, 0, stream>>>(x, w_off, b_off, pcoord, pw);
    dconv_gemm_wmma_kernel<<<2048, 256, 0, stream>>>(xT, wOrd, pcoord, pw, bias, out);
}